// ProposalGenerator_57166014709945
// MI455X (gfx1250) — compile-verified
//
#include <hip/hip_runtime.h>
#include <hip/hip_bf16.h>
#include <math.h>

#ifndef __has_builtin
#define __has_builtin(x) 0
#endif

#define K_PRE   6000
#define K_PAD   8192
#define POST_K  1000
#define NBINS   8192
#define NMS_TH  0.7f
#define NEGV    (-1000000000.0f)
#define MIN_SZ  16.0f

// ---------------- gfx1250-specific data movement helpers ----------------
#if defined(__AMDGCN__)
  #if __has_builtin(__builtin_amdgcn_global_load_async_to_lds_b128)
    #define HAVE_ASYNC_B128 1
  #endif
  #if __has_builtin(__builtin_amdgcn_s_wait_asynccnt)
    #define HAVE_WAIT_ASYNC 1
  #endif
#endif

typedef int pg_v4i __attribute__((vector_size(16)));

__device__ __forceinline__ void async_copy16(const void* g, void* l) {
#if defined(HAVE_ASYNC_B128)
  // GLOBAL_LOAD_ASYNC_TO_LDS_B128: per-lane 16B gather straight into LDS,
  // tracked with ASYNCcnt (no VGPR round trip).
  __builtin_amdgcn_global_load_async_to_lds_b128(
      (__attribute__((address_space(1))) pg_v4i*)(void*)g,
      (__attribute__((address_space(3))) pg_v4i*)l, 0, 0);
#else
  *(float4*)l = *(const float4*)g;
#endif
}

__device__ __forceinline__ void wait_async0() {
#if defined(HAVE_WAIT_ASYNC)
  __builtin_amdgcn_s_wait_asynccnt(0);
#elif defined(__AMDGCN__)
  asm volatile("s_wait_asynccnt 0" ::: "memory");
#endif
}

// ---------------- kernel 0: zero hist/counters, init topk pads ----------------
__global__ void pg_init_kernel(unsigned* hist, unsigned* cntA, unsigned* cntE,
                               float* topk_score, int* topk_idx, int B) {
  int i = blockIdx.x * blockDim.x + threadIdx.x;
  int nh = B * NBINS;
  if (i < nh) hist[i] = 0u;
  if (i < B) { cntA[i] = 0u; cntE[i] = 0u; }
  int nt = B * K_PAD;
  if (i < nt) { topk_score[i] = NEGV; topk_idx[i] = -1; }
}

// ---------------- kernel 1: decode boxes + score + score histogram ----------------
// Bandwidth-dominant pass (~100MB): all row data moved as B128 vector loads.
__global__ void pg_decode_kernel(const float4* __restrict__ anchors4,
                                 const int* __restrict__ bidx,
                                 const int* __restrict__ img_sizes,
                                 const float* __restrict__ logits,
                                 const float4* __restrict__ deltas4,
                                 float4* __restrict__ boxes,
                                 float* __restrict__ score,
                                 unsigned* __restrict__ hist, int N) {
  int i = blockIdx.x * blockDim.x + threadIdx.x;
  if (i >= N) return;
  __builtin_prefetch(&anchors4[(size_t)i + 4096], 0, 1);  // global_prefetch_b8
  __builtin_prefetch(&deltas4[(size_t)i + 4096], 0, 1);

  float4 a = anchors4[i];   // global_load_b128
  float4 t = deltas4[i];    // global_load_b128
  float aw = a.z - a.x, ah = a.w - a.y;
  float acx = a.x + 0.5f*aw, acy = a.y + 0.5f*ah;
  float px = acx + t.x*aw, py = acy + t.y*ah;
  float pw = aw * expf(fminf(fmaxf(t.z, -10.f), 10.f));
  float ph = ah * expf(fminf(fmaxf(t.w, -10.f), 10.f));
  int b = bidx[i];
  float H = (float)img_sizes[b*2+0], W = (float)img_sizes[b*2+1];
  float x1 = fminf(fmaxf(px - 0.5f*pw, 0.f), W - 1.f);
  float y1 = fminf(fmaxf(py - 0.5f*ph, 0.f), H - 1.f);
  float x2 = fminf(fmaxf(px + 0.5f*pw, 0.f), W - 1.f);
  float y2 = fminf(fmaxf(py + 0.5f*ph, 0.f), H - 1.f);
  boxes[i] = make_float4(x1, y1, x2, y2);  // global_store_b128
  bool valid = (x2 - x1 >= MIN_SZ) && (y2 - y1 >= MIN_SZ);
  float s = NEGV;
  if (valid) {
    s = 1.f / (1.f + expf(-logits[i]));
    int bin = (int)(s * (float)NBINS);
    bin = bin < 0 ? 0 : (bin > NBINS - 1 ? NBINS - 1 : bin);
    atomicAdd(&hist[b * NBINS + bin], 1u);
  }
  score[i] = s;
}

// ---------------- kernel 2: find per-image threshold bin ----------------
__global__ void pg_thresh_kernel(const unsigned* __restrict__ hist,
                                 unsigned* __restrict__ threshBin,
                                 unsigned* __restrict__ aboveCnt) {
  const int b = blockIdx.x;
  const int tid = threadIdx.x;        // 256 threads
  const int per = NBINS / 256;        // 32 bins per thread
  __shared__ unsigned chunk[256];
  unsigned s = 0;
  for (int k = 0; k < per; ++k) s += hist[b * NBINS + tid * per + k];
  chunk[tid] = s;
  __syncthreads();
  if (tid == 0) {
    unsigned cum = 0;
    int c = 255;
    for (; c >= 0; --c) {
      if (cum + chunk[c] >= (unsigned)K_PRE) break;
      cum += chunk[c];
    }
    unsigned tb, above;
    if (c >= 0) {
      int base = b * NBINS + c * per;
      int bin = per - 1;
      for (; bin > 0; --bin) {
        unsigned h = hist[base + bin];
        if (cum + h >= (unsigned)K_PRE) break;
        cum += h;
      }
      tb = (unsigned)(c * per + bin);
      above = cum;                       // count strictly above threshold bin
    } else {
      tb = 0u;                           // fewer than K_PRE valid boxes total
      above = cum - hist[b * NBINS + 0];
    }
    threshBin[b] = tb;
    aboveCnt[b] = above;
  }
}

// ---------------- kernel 3: compact candidates >= threshold bin ----------------
__global__ void pg_compact_kernel(const float* __restrict__ score,
                                  const int* __restrict__ bidx,
                                  const unsigned* __restrict__ threshBin,
                                  const unsigned* __restrict__ aboveCnt,
                                  unsigned* __restrict__ cntA, unsigned* __restrict__ cntE,
                                  float* __restrict__ topk_score,
                                  int* __restrict__ topk_idx, int N) {
  int i = blockIdx.x * blockDim.x + threadIdx.x;
  if (i >= N) return;
  float s = score[i];
  if (s <= 0.5f * NEGV) return;
  int b = bidx[i];
  int bin = (int)(s * (float)NBINS);
  bin = bin < 0 ? 0 : (bin > NBINS - 1 ? NBINS - 1 : bin);
  int tb = (int)threshBin[b];
  int slot = -1;
  if (bin > tb) {
    slot = (int)atomicAdd(&cntA[b], 1u);           // guaranteed < K_PRE
  } else if (bin == tb) {
    slot = (int)aboveCnt[b] + (int)atomicAdd(&cntE[b], 1u);  // whole tie-bin kept
  }
  if (slot >= 0 && slot < K_PAD) {
    topk_score[(size_t)b * K_PAD + slot] = s;
    topk_idx[(size_t)b * K_PAD + slot] = i;
  }
}

// ---------------- kernel 4: per-image bitonic sort (score desc, idx asc) ----------------
__global__ void pg_sort_kernel(float* __restrict__ topk_score, int* __restrict__ topk_idx) {
  const int b = blockIdx.x;
  const int tid = threadIdx.x, nthr = blockDim.x;  // 1024 thr = 32 waves (wave32)
  extern __shared__ char smem[];
  float* sc = (float*)smem;            // K_PAD
  int* id = (int*)(sc + K_PAD);        // K_PAD
  for (int j = tid; j < K_PAD; j += nthr) {
    sc[j] = topk_score[(size_t)b * K_PAD + j];
    id[j] = topk_idx[(size_t)b * K_PAD + j];
  }
  __syncthreads();
  for (int k = 2; k <= K_PAD; k <<= 1) {
    for (int j = k >> 1; j > 0; j >>= 1) {
      for (int p = tid; p < K_PAD / 2; p += nthr) {
        int i0 = ((p / j) * (j << 1)) + (p % j);
        int i1 = i0 + j;
        bool desc = ((i0 & k) == 0);
        float s0 = sc[i0], s1 = sc[i1];
        int a0 = id[i0], a1 = id[i1];
        bool firstBeats = (s0 > s1) || ((s0 == s1) && (a0 < a1));
        bool doSwap = desc ? !firstBeats : firstBeats;
        if (doSwap) { sc[i0] = s1; sc[i1] = s0; id[i0] = a1; id[i1] = a0; }
      }
      __syncthreads();
    }
  }
  for (int j = tid; j < K_PAD; j += nthr) {
    topk_score[(size_t)b * K_PAD + j] = sc[j];
    topk_idx[(size_t)b * K_PAD + j] = id[j];
  }
}

// ---------------- kernel 5: greedy NMS, one workgroup per image, all-LDS ----------------
// 6000 boxes (96KB AoS float4) + areas + flags + pick log = ~148KB LDS: only
// possible because CDNA5 gives 320KB LDS per WGP. Boxes are gathered with
// GLOBAL_LOAD_ASYNC_TO_LDS_B128 (ASYNCcnt path).
__global__ void pg_nms_kernel(const float4* __restrict__ boxes,
                              const float* __restrict__ topk_score,
                              const int* __restrict__ topk_idx,
                              int* __restrict__ nms_idx, int* __restrict__ nms_ok) {
  const int b = blockIdx.x;
  const int tid = threadIdx.x;
  const int nthr = blockDim.x;
  extern __shared__ char smem[];
  float4* bx   = (float4*)smem;                  // K_PRE boxes
  float* area  = (float*)(bx + K_PRE);           // K_PRE areas
  unsigned* sup = (unsigned*)(area + K_PRE);     // K_PRE suppression flags
  int* picks    = (int*)(sup + K_PRE);           // POST_K
  unsigned* oks = (unsigned*)(picks + POST_K);   // POST_K
  __shared__ int sPos, sPick;

  // Stage sorted top boxes into LDS (async gather per lane).
  for (int j = tid; j < K_PRE; j += nthr) {
    int gi = topk_idx[(size_t)b * K_PAD + j];
    float sc = topk_score[(size_t)b * K_PAD + j];
    if (gi >= 0 && sc > 0.5f * NEGV) {
      async_copy16(&boxes[gi], &bx[j]);
      sup[j] = 0u;
    } else {
      bx[j] = make_float4(0.f, 0.f, 0.f, 0.f);
      sup[j] = 1u;                               // invalid -> pre-suppressed
    }
  }
  wait_async0();
  __syncthreads();
  for (int j = tid; j < K_PRE; j += nthr) {
    float4 v = bx[j];
    area[j] = (v.z - v.x) * (v.w - v.y);
  }
  if (tid == 0) sPos = 0;
  __syncthreads();

  // Sequential-scan greedy NMS == argmax greedy on the descending-sorted list.
  for (int it = 0; it < POST_K; ++it) {
    if (tid == 0) {
      int p = sPos;
      while (p < K_PRE && sup[p]) ++p;
      sPick = p;
      if (p < K_PRE) { sup[p] = 1u; sPos = p + 1; picks[it] = p; oks[it] = 1u; }
      else           { sPos = p;    picks[it] = 0; oks[it] = 0u; }
    }
    __syncthreads();
    int p = sPick;
    if (p < K_PRE) {
      float4 pb = bx[p];
      float pa = area[p];
      for (int j = p + 1 + tid; j < K_PRE; j += nthr) {
        if (!sup[j]) {
          float4 q = bx[j];
          float ix1 = fmaxf(pb.x, q.x), iy1 = fmaxf(pb.y, q.y);
          float ix2 = fminf(pb.z, q.z), iy2 = fminf(pb.w, q.w);
          float inter = fmaxf(ix2 - ix1, 0.f) * fmaxf(iy2 - iy1, 0.f);
          float iou = inter / (pa + area[j] - inter + 1e-9f);
          if (iou > NMS_TH) sup[j] = 1u;
        }
      }
    }
    __syncthreads();
  }

  for (int it = tid; it < POST_K; it += nthr) {
    nms_idx[b * POST_K + it] = picks[it];
    nms_ok[b * POST_K + it] = (int)oks[it];
  }
}

// ---------------- kernel 6: gather final outputs ----------------
__global__ void pg_gather_kernel(const float4* __restrict__ boxes,
                                 const float* __restrict__ logits,
                                 const float4* __restrict__ deltas4,
                                 const int* __restrict__ topk_idx,
                                 const int* __restrict__ nms_idx,
                                 const int* __restrict__ nms_ok,
                                 float* __restrict__ out, int B) {
  int e = blockIdx.x * blockDim.x + threadIdx.x;
  const int total = B * POST_K;
  if (e >= total) return;
  int b = e / POST_K;
  int ok = nms_ok[e];
  int pick = nms_idx[e];
  int sel = topk_idx[(size_t)b * K_PAD + pick];
  float4 bxv = make_float4(0.f, 0.f, 0.f, 0.f);
  float4 dv  = make_float4(0.f, 0.f, 0.f, 0.f);
  float obj = 0.f;
  if (ok && sel >= 0) {
    bxv = boxes[sel];      // global_load_b128
    dv  = deltas4[sel];    // global_load_b128
    obj = logits[sel];
  }
  float* props = out;                        // [total*4]
  float* obidx = out + (size_t)total * 4;    // [total]
  float* oobj  = out + (size_t)total * 5;    // [total]
  float* odel  = out + (size_t)total * 6;    // [total*4]
  float* okeep = out + (size_t)total * 10;   // [total]
  ((float4*)props)[e] = bxv;                 // global_store_b128
  obidx[e] = ok ? (float)b : -1.0f;
  oobj[e] = obj;
  ((float4*)odel)[e] = dv;                   // global_store_b128
  okeep[e] = ok ? 1.0f : 0.0f;
}

// ---------------- host launcher ----------------
extern "C" void kernel_launch(void* const* d_in, const int* in_sizes, int n_in,
                              void* d_out, int out_size, void* d_ws, size_t ws_size,
                              hipStream_t stream) {
  const float4* anchors4     = (const float4*)d_in[0];
  const int*   batch_indices = (const int*)d_in[1];
  const int*   image_sizes   = (const int*)d_in[2];
  const float* logits        = (const float*)d_in[3];
  const float4* deltas4      = (const float4*)d_in[4];
  const int N = in_sizes[0] / 4;
  const int B = in_sizes[2] / 2;

  char* ws = (char*)d_ws;
  size_t off = 0;
  float4*   wsBoxes = (float4*)(ws + off);  off += (size_t)N * 16;
  float*    wsScore = (float*)(ws + off);   off += (size_t)N * 4;
  unsigned* wsHist  = (unsigned*)(ws + off); off += (size_t)B * NBINS * 4;
  unsigned* wsTB    = (unsigned*)(ws + off); off += (size_t)B * 4;
  unsigned* wsAbove = (unsigned*)(ws + off); off += (size_t)B * 4;
  unsigned* wsCntA  = (unsigned*)(ws + off); off += (size_t)B * 4;
  unsigned* wsCntE  = (unsigned*)(ws + off); off += (size_t)B * 4;
  float*    wsTS    = (float*)(ws + off);   off += (size_t)B * K_PAD * 4;
  int*      wsTI    = (int*)(ws + off);     off += (size_t)B * K_PAD * 4;
  int*      wsNI    = (int*)(ws + off);     off += (size_t)B * POST_K * 4;
  int*      wsNO    = (int*)(ws + off);     off += (size_t)B * POST_K * 4;

  const int T = 256;
  int nInit = B * NBINS > B * K_PAD ? B * NBINS : B * K_PAD;
  pg_init_kernel<<<(nInit + T - 1) / T, T, 0, stream>>>(wsHist, wsCntA, wsCntE, wsTS, wsTI, B);
  pg_decode_kernel<<<(N + T - 1) / T, T, 0, stream>>>(anchors4, batch_indices, image_sizes,
                                                      logits, deltas4, wsBoxes, wsScore,
                                                      wsHist, N);
  pg_thresh_kernel<<<B, 256, 0, stream>>>(wsHist, wsTB, wsAbove);
  pg_compact_kernel<<<(N + T - 1) / T, T, 0, stream>>>(wsScore, batch_indices, wsTB, wsAbove,
                                                       wsCntA, wsCntE, wsTS, wsTI, N);
  pg_sort_kernel<<<B, 1024, (size_t)K_PAD * 8, stream>>>(wsTS, wsTI);
  size_t nmsLds = (size_t)K_PRE * 16 + (size_t)K_PRE * 4 + (size_t)K_PRE * 4
                + (size_t)POST_K * 4 + (size_t)POST_K * 4;   // ~148 KB < 320 KB WGP LDS
  pg_nms_kernel<<<B, 1024, nmsLds, stream>>>(wsBoxes, wsTS, wsTI, wsNI, wsNO);
  pg_gather_kernel<<<(B * POST_K + T - 1) / T, T, 0, stream>>>(wsBoxes, logits, deltas4,
                                                               wsTI, wsNI, wsNO,
                                                               (float*)d_out, B);
}